// Interp23Tap_14276471292324
// MI455X (gfx1250) — compile-verified
//
#include <hip/hip_runtime.h>
#include <hip/hip_bf16.h>

// ---------------------------------------------------------------------------
// 4x polyphase upsampler (two 2x half-band stages), MI455X / gfx1250.
// Memory-bound (~223 MB min traffic, ~1.6 GFLOP): optimize data movement.
//  - Stage 1: TDM (tensor_load_to_lds) DMA of input strips into LDS,
//             polyphase W pass -> LDS, circular H pass -> B in workspace.
//  - Stage 2: tiled fused W+H from B, NT (non-temporal) output stores so the
//             210 MB output stream does not thrash L2 (B stays resident).
// Half-band taps: even taps zero except unity center -> 12 FMAs per sample.
// ---------------------------------------------------------------------------

#define C1f  ( 0.610668182370f)
#define C3f  (-0.145397186478f)
#define C5f  ( 0.043619155884f)
#define C7f  (-0.010385513306f)
#define C9f  ( 0.001615524292f)
#define C11f (-0.000120162964f)

typedef unsigned int v4u __attribute__((ext_vector_type(4)));
typedef int          v8i __attribute__((ext_vector_type(8)));
typedef int          v4i __attribute__((ext_vector_type(4)));

// Build a 2D TDM descriptor (CDNA5 ISA ch.8) and issue TENSOR_LOAD_TO_LDS.
// tile: tile_w x tile_h f32 elements, row stride = stride_elems, packed
// contiguously into LDS at lds_dst. clang-23 / therock 6-arg builtin form:
// (group0 v4u, group1 v8i, group2 v4i, group3 v4i, extra v8i, cpol i32).
__device__ __forceinline__ void tdm_load_2d_f32(const void* gsrc, void* lds_dst,
                                                unsigned tile_w, unsigned tile_h,
                                                unsigned stride_elems,
                                                unsigned tensor_w, unsigned tensor_h) {
  unsigned long long ga = (unsigned long long)gsrc;
  unsigned lds = (unsigned)(unsigned long long)lds_dst; // low 32 bits = LDS offset
  v4u g0;
  g0.x = 1u;                                            // count=1 (valid user D#)
  g0.y = lds;                                           // lds_addr [63:32]
  g0.z = (unsigned)ga;                                  // global_addr [95:64]
  g0.w = (unsigned)((ga >> 32) & 0x1FFFFFFull) | (2u << 30); // addr[56:32], type=2
  v8i g1;
  g1[0] = (int)(2u << 16);                              // data_size=2 (4 bytes)
  g1[1] = (int)((tensor_w & 0xFFFFu) << 16);            // tensor_dim0 lo
  g1[2] = (int)((tensor_w >> 16) | ((tensor_h & 0xFFFFu) << 16));
  g1[3] = (int)((tensor_h >> 16) | ((tile_w & 0xFFFFu) << 16)); // tile_dim0
  g1[4] = (int)(tile_h & 0xFFFFu);                      // tile_dim1, tile_dim2=0
  g1[5] = (int)stride_elems;                            // tensor_dim0_stride lo
  g1[6] = 0;
  g1[7] = 0;
  v4i z4 = {0, 0, 0, 0};                                // 2D: groups 2/3 unused
  v8i z8 = {0, 0, 0, 0, 0, 0, 0, 0};
  __builtin_amdgcn_tensor_load_to_lds(g0, g1, z4, z4, z8, 0);
}

// ---------------------------------------------------------------------------
// Kernel 1: x(160x160) -> B(320x320) per plane. grid=(8 strips, 128 planes).
// Strip s covers B columns [40s, 40s+39]; needs x cols [20s-6, 20s+24],
// staged as a fully in-bounds 32-wide tile starting at clamp(20s-6, 0, 128).
// ---------------------------------------------------------------------------
__global__ __launch_bounds__(256) void Interp23_stage1(const float* __restrict__ x,
                                                       float* __restrict__ B) {
  const int s     = blockIdx.x;
  const int plane = blockIdx.y;
  __shared__ float xs[160 * 32];   // 20.0 KB input tile
  __shared__ float as[160 * 40];   // 25.0 KB width-pass intermediate

  const int tile_start = min(max(s * 20 - 6, 0), 128);
  const float* xp = x + (size_t)plane * (160 * 160) + tile_start;

  if (threadIdx.x == 0) {
    tdm_load_2d_f32(xp, xs, /*tile_w=*/32, /*tile_h=*/160, /*stride=*/160,
                    /*tensor_w=*/160, /*tensor_h=*/160);
    __builtin_amdgcn_s_wait_tensorcnt(0);   // TDM tracked by TENSORcnt
  }
  __syncthreads();

  const float t[12] = {C11f, C9f, C7f, C5f, C3f, C1f, C1f, C3f, C5f, C7f, C9f, C11f};
  const float P[6]  = {C1f + C3f + C5f + C7f + C9f + C11f,
                       C3f + C5f + C7f + C9f + C11f,
                       C5f + C7f + C9f + C11f,
                       C7f + C9f + C11f,
                       C9f + C11f,
                       C11f};

  // Width pass (edge pad on zero-inserted grid, samples at odd indices):
  //  A[i,2j+1] = x[i,j] (+ right-edge replication corrections)
  //  A[i,2j]   = sum_m t[m+6] * gx(j+m),  gx: <0 -> 0, >159 -> x[159]
  for (int idx = threadIdx.x; idx < 160 * 40; idx += 256) {
    const int i  = idx / 40;
    const int wl = idx - i * 40;
    const int w  = s * 40 + wl;
    const int j  = w >> 1;
    const float* xr = xs + i * 32;
    float v;
    if (w & 1) {
      v = xr[j - tile_start];
      if (j >= 154) v = fmaf(P[159 - j], xr[159 - tile_start], v);
    } else {
      v = 0.0f;
#pragma unroll
      for (int m = -6; m <= 5; ++m) {
        const int tt = j + m;
        const float xv = (tt < 0) ? 0.0f : xr[min(tt, 159) - tile_start];
        v = fmaf(t[m + 6], xv, v);
      }
    }
    as[i * 40 + wl] = v;
  }
  __syncthreads();

  // Height pass (circular): B[2i+1]=A[i]; B[2i]=sum_m t[m+6]*A[(i+m) mod 160]
  float* Bp = B + (size_t)plane * (320 * 320);
  for (int idx = threadIdx.x; idx < 320 * 40; idx += 256) {
    const int h  = idx / 40;
    const int wl = idx - h * 40;
    const int i  = h >> 1;
    float v;
    if (h & 1) {
      v = as[i * 40 + wl];
    } else {
      v = 0.0f;
#pragma unroll
      for (int m = -6; m <= 5; ++m) {
        int r = i + m;
        r = (r < 0) ? r + 160 : (r >= 160 ? r - 160 : r);
        v = fmaf(t[m + 6], as[r * 40 + wl], v);
      }
    }
    Bp[(size_t)h * 320 + s * 40 + wl] = v;
  }
}

// ---------------------------------------------------------------------------
// Kernel 2: B(320x320) -> O(640x640) per plane, fused W2+H2.
// grid=(5 col tiles, 20 row tiles, 128 planes). Output tile 32 rows x 128 cols.
// ---------------------------------------------------------------------------
__global__ __launch_bounds__(256) void Interp23_stage2(const float* __restrict__ B,
                                                       float* __restrict__ O) {
  const int c0    = blockIdx.x * 64;   // B-space column base (64 B cols = 128 C cols / tile)
  const int r0    = blockIdx.y * 16;   // C-space row base (16 C rows / tile)
  const int plane = blockIdx.z;

  __shared__ float bt[27][76];     // B tile: 27 rows x 75 cols (+1 pad)
  __shared__ float ct[27][128];    // C tile (stage-2 width-pass result)

  const float* Bp = B + (size_t)plane * (320 * 320);

  // Load B tile. Rows wrap mod 320 (circular H). Columns use stage-2 width
  // edge semantics baked in: col<0 -> B[row][0] (clamp), col>319 -> 0.
  for (int idx = threadIdx.x; idx < 27 * 75; idx += 256) {
    const int lr = idx / 75;
    const int lb = idx - lr * 75;
    int gr = r0 - 5 + lr;
    gr = (gr < 0) ? gr + 320 : (gr >= 320 ? gr - 320 : gr);
    const int gc = c0 - 5 + lb;
    float v = 0.0f;
    if (gc <= 319) v = Bp[(size_t)gr * 320 + max(gc, 0)];
    bt[lr][lb] = v;
  }
  __syncthreads();

  const float t[12] = {C11f, C9f, C7f, C5f, C3f, C1f, C1f, C3f, C5f, C7f, C9f, C11f};
  const float P[6]  = {C1f + C3f + C5f + C7f + C9f + C11f,
                       C3f + C5f + C7f + C9f + C11f,
                       C5f + C7f + C9f + C11f,
                       C7f + C9f + C11f,
                       C9f + C11f,
                       C11f};

  // Width pass: C[r,2j] = B[r,j] (+ left-edge clamp corrections P[j]*B[r,0]);
  //             C[r,2j+1] = sum_{m=-5..6} t[m+5] * gc(j+m)
  for (int idx = threadIdx.x; idx < 27 * 64; idx += 256) {
    const int lr = idx / 64;
    const int jl = idx - lr * 64;
    const int j  = c0 + jl;
    float ve = bt[lr][jl + 5];
    if (j <= 5) ve = fmaf(P[j], bt[lr][5], ve);
    float vo = 0.0f;
#pragma unroll
    for (int m = -5; m <= 6; ++m)
      vo = fmaf(t[m + 5], bt[lr][jl + 5 + m], vo);
    ct[lr][2 * jl]     = ve;
    ct[lr][2 * jl + 1] = vo;
  }
  __syncthreads();

  // Height pass (circular, handled by wrapped tile rows):
  //  O[2r] = C[r];  O[2r+1] = sum_{m=-5..6} t[m+5] * C[r+m]
  float* Op = O + (size_t)plane * (640 * 640);
  for (int idx = threadIdx.x; idx < 32 * 128; idx += 256) {
    const int hl  = idx / 128;
    const int col = idx - hl * 128;
    const int ho  = 2 * r0 + hl;
    float v;
    if ((hl & 1) == 0) {
      v = ct[(hl >> 1) + 5][col];
    } else {
      const int base = ((hl - 1) >> 1) + 5;
      v = 0.0f;
#pragma unroll
      for (int m = -5; m <= 6; ++m)
        v = fmaf(t[m + 5], ct[base + m][col], v);
    }
    // Streaming store: keep the 210 MB output out of L2 (B stays resident).
    __builtin_nontemporal_store(v, &Op[(size_t)ho * 640 + 2 * c0 + col]);
  }
}

extern "C" void kernel_launch(void* const* d_in, const int* in_sizes, int n_in,
                              void* d_out, int out_size, void* d_ws, size_t ws_size,
                              hipStream_t stream) {
  const float* x = (const float*)d_in[0];
  float* O = (float*)d_out;
  float* B = (float*)d_ws;  // 128 * 320 * 320 * 4 = 52,428,800 bytes of scratch

  Interp23_stage1<<<dim3(8, 128), 256, 0, stream>>>(x, B);
  Interp23_stage2<<<dim3(5, 20, 128), 256, 0, stream>>>(B, O);
}